// ControlFlowScan_46308337386059
// MI455X (gfx1250) — compile-verified
//
#include <hip/hip_runtime.h>

// Column-sum of an N x D fp32 matrix (N=8192, D=4096): out[d] = sum_n x[n][d].
// Memory-bound: 128 MiB streamed once -> ~5.8 us floor at 23.3 TB/s.
// Pass 1: NT b128 streaming loads, partials per row-tile into d_ws (deterministic).
// Pass 2: reduce 128 x D partials with V_WMMA_F32_16X16X4_F32 using an all-ones
//         A matrix (D[m][n] = sum_k B[k][n]); K-permutation-invariant, so only
//         the documented N-per-lane layout matters.

typedef __attribute__((ext_vector_type(4))) float f32x4;
typedef __attribute__((ext_vector_type(2))) float v2f;
typedef __attribute__((ext_vector_type(8))) float v8f;

#define COLS_PER_THREAD 4
#define BLOCK_THREADS   256
#define ROW_TILES       128

// ---------------- Pass 1: streaming partial column sums ----------------
__global__ void colsum_partial_kernel(const float* __restrict__ x,
                                      float* __restrict__ ws,
                                      int N, int D, int rows_per_tile) {
    int col = (blockIdx.x * BLOCK_THREADS + threadIdx.x) * COLS_PER_THREAD;
    if (col >= D) return;

    int r0 = blockIdx.y * rows_per_tile;
    int r1 = r0 + rows_per_tile;
    if (r1 > N) r1 = N;
    if (r0 >= r1) return;

    if (col + COLS_PER_THREAD <= D) {
        f32x4 a0 = {0.f, 0.f, 0.f, 0.f};
        f32x4 a1 = a0, a2 = a0, a3 = a0;
        const float* p = x + (size_t)r0 * (size_t)D + (size_t)col;
        int r = r0;
        // 4-deep unroll: independent accumulators, 4 b128 NT loads in flight.
        for (; r + 4 <= r1; r += 4) {
            f32x4 v0 = __builtin_nontemporal_load((const f32x4*)(p));
            f32x4 v1 = __builtin_nontemporal_load((const f32x4*)(p + (size_t)D));
            f32x4 v2 = __builtin_nontemporal_load((const f32x4*)(p + 2u * (size_t)D));
            f32x4 v3 = __builtin_nontemporal_load((const f32x4*)(p + 3u * (size_t)D));
            a0 += v0; a1 += v1; a2 += v2; a3 += v3;
            p += 4u * (size_t)D;
        }
        for (; r < r1; ++r) {
            a0 += __builtin_nontemporal_load((const f32x4*)(p));
            p += (size_t)D;
        }
        f32x4 s = (a0 + a1) + (a2 + a3);
        *(f32x4*)(ws + (size_t)blockIdx.y * (size_t)D + (size_t)col) = s;
    } else {
        // Scalar column tail (not hit for D=4096).
        for (int c = col; c < D; ++c) {
            float s = 0.f;
            for (int r = r0; r < r1; ++r)
                s += x[(size_t)r * (size_t)D + (size_t)c];
            ws[(size_t)blockIdx.y * (size_t)D + (size_t)c] = s;
        }
    }
}

// ---------------- Pass 2: WMMA reduction of P x D partials ----------------
// One wave32 per 16-column tile. A = ones(16x4) => every row of D (and hence
// every lane's c[0]) holds sum_k B[k][n] for n = lane & 15. Chain C across
// P/4 WMMAs to sum all P partial rows.
__global__ __launch_bounds__(32)
void colsum_final_wmma_kernel(const float* __restrict__ ws,
                              float* __restrict__ out,
                              int P, int D) {
    const int lane = threadIdx.x;          // 0..31
    const int col_base = blockIdx.x * 16;
    if (col_base >= D) return;             // block-uniform

    if (col_base + 16 <= D) {              // block-uniform: EXEC all-1s for WMMA
        const int half = lane >> 4;        // 0: rows k+0,k+1 ; 1: rows k+2,k+3
        const int n    = lane & 15;
        const int col  = col_base + n;

        v2f aone; aone.x = 1.0f; aone.y = 1.0f;   // all-ones A fragment
        v8f c = {};

        const int P4 = P & ~3;
        for (int k = 0; k < P4; k += 4) {
            const float* q = ws + (size_t)(k + 2 * half) * (size_t)D + (size_t)col;
            v2f b;
            b.x = q[0];                    // B row (k + 2*half)     , col n
            b.y = q[(size_t)D];            // B row (k + 2*half + 1) , col n
            // D = A(ones) x B + C : accumulates column sums of 4 partial rows.
            c = __builtin_amdgcn_wmma_f32_16x16x4_f32(
                    /*neg_a=*/false, aone,
                    /*neg_b=*/false, b,
                    /*c_mod=*/(short)0, c,
                    /*reuse_a=*/false, /*reuse_b=*/false);
        }

        float s = c[0];                    // every lane: colsum for its n
        for (int k = P4; k < P; ++k)       // leftover rows (none for P=128)
            s += ws[(size_t)k * (size_t)D + (size_t)col];

        if (half == 0)                     // lanes 0..15 store the 16 columns
            out[col] = s;
    } else {
        // Scalar tail tile (not hit for D=4096).
        int col = col_base + lane;
        if (lane < 16 && col < D) {
            float s = 0.f;
            for (int k = 0; k < P; ++k)
                s += ws[(size_t)k * (size_t)D + (size_t)col];
            out[col] = s;
        }
    }
}

// ---------------- Fallback: single pass (if d_ws too small) ----------------
__global__ void colsum_single_kernel(const float* __restrict__ x,
                                     float* __restrict__ out,
                                     int N, int D) {
    int col = (blockIdx.x * BLOCK_THREADS + threadIdx.x) * COLS_PER_THREAD;
    if (col >= D) return;
    if (col + COLS_PER_THREAD <= D) {
        f32x4 a = {0.f, 0.f, 0.f, 0.f};
        const float* p = x + (size_t)col;
        for (int r = 0; r < N; ++r) {
            a += __builtin_nontemporal_load((const f32x4*)(p));
            p += (size_t)D;
        }
        *(f32x4*)(out + col) = a;
    } else {
        for (int c = col; c < D; ++c) {
            float s = 0.f;
            for (int r = 0; r < N; ++r)
                s += x[(size_t)r * (size_t)D + (size_t)c];
            out[c] = s;
        }
    }
}

extern "C" void kernel_launch(void* const* d_in, const int* in_sizes, int n_in,
                              void* d_out, int out_size, void* d_ws, size_t ws_size,
                              hipStream_t stream) {
    const float* x = (const float*)d_in[0];
    float* out = (float*)d_out;

    const int D = out_size;                       // 4096
    const int N = (D > 0) ? (in_sizes[0] / D) : 0; // 8192

    const int cols_per_block = BLOCK_THREADS * COLS_PER_THREAD; // 1024
    const int col_blocks = (D + cols_per_block - 1) / cols_per_block;

    const size_t ws_need = (size_t)ROW_TILES * (size_t)D * sizeof(float); // 2 MiB

    if (ws_size >= ws_need && N >= ROW_TILES) {
        const int rows_per_tile = (N + ROW_TILES - 1) / ROW_TILES; // 64
        float* partials = (float*)d_ws;

        dim3 grid1(col_blocks, ROW_TILES);
        colsum_partial_kernel<<<grid1, BLOCK_THREADS, 0, stream>>>(
            x, partials, N, D, rows_per_tile);

        const int tiles16 = (D + 15) / 16;        // 256 wave-blocks
        colsum_final_wmma_kernel<<<tiles16, 32, 0, stream>>>(
            partials, out, ROW_TILES, D);
    } else {
        colsum_single_kernel<<<col_blocks, BLOCK_THREADS, 0, stream>>>(x, out, N, D);
    }
}